// GCN_24988119728436
// MI455X (gfx1250) — compile-verified
//
#include <hip/hip_runtime.h>

// GCN layer for MI455X (gfx1250, wave32).
// Pipeline: init -> degree (atomic) -> rsqrt -> WMMA f32 GEMM (xw = (x*mask)@W)
//           -> edge scatter (global_atomic_add_f32, L2-resident) -> finalize.
// ws layout: xw [N*128 f32] | deg/dinv [N f32]   (~51.6 MB)

typedef float v2f __attribute__((ext_vector_type(2)));
typedef float v8f __attribute__((ext_vector_type(8)));

#define DD 128
#define NEG_SLOPE 0.2f
#define LDS_STRIDE 132  // 128 + 4 pad: conflict-free ds_load_b64, 16B-aligned rows

__global__ void k_init(float* __restrict__ out, float* __restrict__ deg,
                       int n, int total) {
    int i = blockIdx.x * blockDim.x + threadIdx.x;
    if (i < total) out[i] = 0.0f;
    if (i < n) deg[i] = 1.0f;  // self loop contributes 1 to every dst degree
}

__global__ void k_deg(const int* __restrict__ dst, float* __restrict__ deg, int e) {
    int i = blockIdx.x * blockDim.x + threadIdx.x;
    if (i < e) unsafeAtomicAdd(&deg[dst[i]], 1.0f);
}

__global__ void k_dinv(float* __restrict__ deg, int n) {
    int i = blockIdx.x * blockDim.x + threadIdx.x;
    if (i < n) deg[i] = rsqrtf(deg[i]);  // deg >= 1 always (self loops)
}

// xw = (x * mask) @ W using V_WMMA_F32_16X16X4_F32.
// Block: 256 threads = 8 waves. Block covers 64 rows; wave w owns cols [16w,16w+16).
// B panel (32 k-steps) lives in 64 VGPRs per wave, loaded once.
__global__ __launch_bounds__(256)
void k_gemm(const float* __restrict__ x, const float* __restrict__ mask,
            const float* __restrict__ W, float* __restrict__ xw, int n) {
    __shared__ float As[64 * LDS_STRIDE];

    const int tid  = threadIdx.x;
    const int row0 = blockIdx.x * 64;

    // Stage A = x*mask for 64 rows, coalesced float4 loads, zero-pad OOB rows.
    #pragma unroll
    for (int it = 0; it < 8; ++it) {
        int slot = it * 256 + tid;      // 2048 float4 slots = 64 rows * 32
        int r    = slot >> 5;
        int c4   = slot & 31;
        int grow = row0 + r;
        float4 v = make_float4(0.f, 0.f, 0.f, 0.f);
        if (grow < n) {
            const float4 xv = ((const float4*)(x    + (size_t)grow * DD))[c4];
            const float4 mv = ((const float4*)(mask + (size_t)grow * DD))[c4];
            v = make_float4(xv.x * mv.x, xv.y * mv.y, xv.z * mv.z, xv.w * mv.w);
        }
        *(float4*)&As[r * LDS_STRIDE + c4 * 4] = v;  // row start 528B -> 16B aligned
    }
    __syncthreads();

    const int wv   = tid >> 5;            // wave id == N-tile index j
    const int lane = tid & 31;
    const int lm   = lane & 15;
    const int kh   = (lane >> 4) << 1;    // k-half: 0 (lanes 0-15) or 2 (lanes 16-31)
    const int col  = wv * 16 + lm;

    // Preload B fragments: bf[kk] = { W[k][col], W[k+1][col] }, k = 4*kk + kh
    v2f bf[32];
    #pragma unroll
    for (int kk = 0; kk < 32; ++kk) {
        int k = kk * 4 + kh;
        bf[kk].x = W[(size_t)k * DD + col];
        bf[kk].y = W[(size_t)(k + 1) * DD + col];
    }

    #pragma unroll
    for (int mt = 0; mt < 4; ++mt) {
        int mrow = row0 + mt * 16;
        if (mrow >= n) break;             // wave-uniform: EXEC stays all-ones
        v8f acc = {};
        #pragma unroll
        for (int kk = 0; kk < 32; ++kk) {
            // A frag: lane holds A[m=lm][k], A[m=lm][k+1] with k = 4*kk + kh
            v2f a = *(const v2f*)&As[(mt * 16 + lm) * LDS_STRIDE + kk * 4 + kh];
            acc = __builtin_amdgcn_wmma_f32_16x16x4_f32(
                false, a, false, bf[kk], (short)0, acc, false, false);
        }
        // C/D layout: VGPR r -> row (mrow + r) lanes 0-15, (mrow + 8 + r) lanes 16-31
        int rbase = mrow + ((lane >> 4) << 3);
        #pragma unroll
        for (int r = 0; r < 8; ++r) {
            int row = rbase + r;
            if (row < n) xw[(size_t)row * DD + col] = acc[r];
        }
    }
}

// One wave per edge; each lane owns 4 consecutive output floats.
// xw and out both fit in the 192MB L2 -> atomics stay on-chip.
__global__ __launch_bounds__(256)
void k_scatter(const int* __restrict__ src, const int* __restrict__ dst,
               const float* __restrict__ xw, const float* __restrict__ dinv,
               float* __restrict__ out, int e) {
    int gid  = blockIdx.x * blockDim.x + threadIdx.x;
    int edge = gid >> 5;
    if (edge >= e) return;
    int lane = gid & 31;
    int s = src[edge];
    int d = dst[edge];
    float nrm = dinv[s] * dinv[d];
    const float4 v = ((const float4*)(xw + (size_t)s * DD))[lane];
    float* o = out + (size_t)d * DD + lane * 4;
    unsafeAtomicAdd(o + 0, v.x * nrm);
    unsafeAtomicAdd(o + 1, v.y * nrm);
    unsafeAtomicAdd(o + 2, v.z * nrm);
    unsafeAtomicAdd(o + 3, v.w * nrm);
}

// out = leaky_relu(out + xw * dinv^2 (self loop) + b)
__global__ void k_final(float* __restrict__ out, const float* __restrict__ xw,
                        const float* __restrict__ dinv, const float* __restrict__ b,
                        int total) {
    int i = blockIdx.x * blockDim.x + threadIdx.x;
    if (i >= total) return;
    int node = i >> 7;
    int dcol = i & 127;
    float di = dinv[node];
    float v  = out[i] + xw[i] * di * di + b[dcol];
    out[i] = (v >= 0.0f) ? v : NEG_SLOPE * v;
}

extern "C" void kernel_launch(void* const* d_in, const int* in_sizes, int n_in,
                              void* d_out, int out_size, void* d_ws, size_t ws_size,
                              hipStream_t stream) {
    const float* x    = (const float*)d_in[0];
    const int*   ei   = (const int*)  d_in[1];
    const float* W    = (const float*)d_in[2];
    const float* b    = (const float*)d_in[3];
    const float* mask = (const float*)d_in[4];

    const int N = in_sizes[0] / DD;
    const int E = in_sizes[1] / 2;
    const int* src = ei;
    const int* dst = ei + E;

    float* out = (float*)d_out;
    float* xw  = (float*)d_ws;
    float* deg = xw + (size_t)N * DD;   // reused as dinv after k_dinv

    const int total = N * DD;
    k_init   <<<(total + 255) / 256, 256, 0, stream>>>(out, deg, N, total);
    k_deg    <<<(E + 255) / 256,     256, 0, stream>>>(dst, deg, E);
    k_dinv   <<<(N + 255) / 256,     256, 0, stream>>>(deg, N);
    k_gemm   <<<(N + 63) / 64,       256, 0, stream>>>(x, mask, W, xw, N);
    {
        long long threads = (long long)E * 32;
        int blocks = (int)((threads + 255) / 256);
        k_scatter<<<blocks, 256, 0, stream>>>(src, dst, xw, deg, out, E);
    }
    k_final  <<<(total + 255) / 256, 256, 0, stream>>>(out, xw, deg, b, total);
}